// Diag_14285061227129
// MI455X (gfx1250) — compile-verified
//
#include <hip/hip_runtime.h>

// out[n, d] = input[n, d] * W[d]   (N = 16384 rows, D = 4096 cols, f32)
//
// Pure HBM-bandwidth-bound stream: ~537 MB total traffic -> ~23 us floor at
// 23.3 TB/s. Strategy:
//   * b128 vectorized, fully coalesced global loads/stores
//   * non-temporal cache policy (input/output are touched exactly once and
//     the 256 MB input exceeds the 192 MB L2 -> don't pollute L2)
//   * W (16 KB) cached in 4 f32x4 registers per thread: with a fixed
//     column-to-thread mapping, each thread needs the same 16 W values for
//     every row, so W costs one load per block, ever.
//   * global_prefetch_b8 of the next row this block will process.

typedef __attribute__((ext_vector_type(4))) float f32x4;

constexpr int kD     = 4096;            // features per row
constexpr int kVecD  = kD / 4;          // 1024 float4 per row
constexpr int kBlock = 256;             // 8 x wave32
constexpr int kJ     = kVecD / kBlock;  // 4 float4 per thread per row

__global__ __launch_bounds__(kBlock)
void diag_scale_kernel(const f32x4* __restrict__ in,
                       const f32x4* __restrict__ Wv,
                       f32x4* __restrict__ out,
                       int rows)
{
    const int tid = threadIdx.x;

    // Register-resident W: thread t always processes column-quads
    // {t, 256+t, 512+t, 768+t} of every row.
    f32x4 w[kJ];
#pragma unroll
    for (int j = 0; j < kJ; ++j)
        w[j] = Wv[j * kBlock + tid];

    for (long long row = blockIdx.x; row < rows; row += gridDim.x) {
        const f32x4* __restrict__ src = in  + row * (long long)kVecD;
        f32x4* __restrict__       dst = out + row * (long long)kVecD;

        // Prefetch the next row this block will handle; tid*kJ stride (64 B
        // per thread) covers the whole 16 KB row across 256 threads.
        const long long nrow = row + gridDim.x;
        if (nrow < rows)
            __builtin_prefetch(in + nrow * (long long)kVecD + tid * kJ, 0, 0);

#pragma unroll
        for (int j = 0; j < kJ; ++j) {
            const int idx = j * kBlock + tid;
            f32x4 v = __builtin_nontemporal_load(&src[idx]);
            v = v * w[j];
            __builtin_nontemporal_store(v, &dst[idx]);
        }
    }
}

extern "C" void kernel_launch(void* const* d_in, const int* in_sizes, int n_in,
                              void* d_out, int out_size, void* d_ws, size_t ws_size,
                              hipStream_t stream)
{
    const float* input = (const float*)d_in[0];
    const float* W     = (const float*)d_in[1];
    float*       out   = (float*)d_out;

    const int rows = in_sizes[0] / kD;          // 16384
    int grid = rows < 4096 ? rows : 4096;       // 4 rows per block

    diag_scale_kernel<<<grid, kBlock, 0, stream>>>(
        (const f32x4*)input, (const f32x4*)W, (f32x4*)out, rows);
}